// model_73950746902698
// MI455X (gfx1250) — compile-verified
//
#include <hip/hip_runtime.h>

typedef __attribute__((ext_vector_type(2))) float v2f;
typedef __attribute__((ext_vector_type(8))) float v8f;

#define NB 32       // batch
#define NT 64       // time steps
#define NH 128      // hidden
#define G3 384      // 3*hidden (r,z,n gates)
#define NTHREADS 512

// One workgroup per GRU. Everything lives in LDS; the 64-step recurrence runs
// in-kernel. Per-step GEMM gh = h @ Whh^T done with V_WMMA_F32_16X16X4_F32.
// Whh staging uses GLOBAL_LOAD_ASYNC_TO_LDS_B64 (CDNA5 async LDS-DMA path):
// the pair-interleaved, bank-conflict-free LDS layout is reachable with pure
// 8-byte scattered copies, so no VGPR round-trip or rearrangement ALU needed.
__global__ __launch_bounds__(NTHREADS) void gru3_kernel(
    const float* __restrict__ bbox, const float* __restrict__ flow, const float* __restrict__ ego,
    const float* __restrict__ Wih_b, const float* __restrict__ Whh_b,
    const float* __restrict__ bih_b, const float* __restrict__ bhh_b,
    const float* __restrict__ Wih_f, const float* __restrict__ Whh_f,
    const float* __restrict__ bih_f, const float* __restrict__ bhh_f,
    const float* __restrict__ Wih_e, const float* __restrict__ Whh_e,
    const float* __restrict__ bih_e, const float* __restrict__ bhh_e,
    float* __restrict__ ws)
{
    // Whh^T staged as k-pairs: element (p=k/2, n, sub=k&1) at [(p*G3+n)*2+sub]
    __shared__ float lds_B[(NH / 2) * G3 * 2]; // 192 KB
    __shared__ float lds_gh[NB * G3];          // 48 KB
    __shared__ float lds_x[NB * NT * 4];       // 32 KB (stride-4 padded inputs)
    __shared__ float lds_h[NB * NH];           // 16 KB
    __shared__ float lds_wih[G3 * 4];          // 6 KB
    __shared__ float lds_bih[G3];
    __shared__ float lds_bhh[G3];

    const int tid = threadIdx.x;
    const int g = blockIdx.x; // 0=bbox, 1=flow(center pixel), 2=ego

    const float* Wih; const float* Whh; const float* bih; const float* bhh; int in_dim;
    if (g == 0)      { Wih = Wih_b; Whh = Whh_b; bih = bih_b; bhh = bhh_b; in_dim = 4; }
    else if (g == 1) { Wih = Wih_f; Whh = Whh_f; bih = bih_f; bhh = bhh_f; in_dim = 2; }
    else             { Wih = Wih_e; Whh = Whh_e; bih = bih_e; bhh = bhh_e; in_dim = 3; }

    // ---- stage Whh into pair-interleaved LDS layout via async LDS-DMA ----
    // Pair p = k/2 in [0,64), column n in [0,384): copy the 8-byte chunk
    //   Whh[n*128 + 2p .. 2p+1]  ->  lds_B[(p*384 + n)*2 .. +1]
    // Both sides are contiguous and 8B aligned, so B64 async copies suffice.
    for (int i = tid; i < (NH / 2) * G3; i += NTHREADS) {       // 48 iters/lane
        int p = i / G3, n = i % G3;
        unsigned lds_off = (unsigned)(uintptr_t)(&lds_B[((p * G3) + n) * 2]);
        unsigned long long ga = (unsigned long long)(uintptr_t)(&Whh[n * NH + 2 * p]);
        asm volatile("global_load_async_to_lds_b64 %0, %1, off"
                     :: "v"(lds_off), "v"(ga) : "memory");
    }
    // ---- stage input weights / biases (tiny; plain loads) ----
    for (int i = tid; i < G3 * in_dim; i += NTHREADS) {
        int j = i / in_dim, d = i % in_dim;
        lds_wih[j * 4 + d] = Wih[i];
    }
    for (int i = tid; i < G3; i += NTHREADS) { lds_bih[i] = bih[i]; lds_bhh[i] = bhh[i]; }
    // ---- stage sequence inputs (flow: gather only the center pixel) ----
    if (g == 0) {
        for (int i = tid; i < NB * NT * 4; i += NTHREADS) lds_x[i] = bbox[i];
    } else if (g == 1) {
        for (int i = tid; i < NB * NT * 2; i += NTHREADS) {
            int bt = i >> 1, c = i & 1;
            size_t idx = ((size_t)(bt * 128 + 64) * 128 + 64) * 2 + c; // flow[b,t,64,64,c]
            lds_x[bt * 4 + c] = flow[idx];
        }
    } else {
        for (int i = tid; i < NB * NT * 3; i += NTHREADS) {
            int bt = i / 3, c = i % 3;
            lds_x[bt * 4 + c] = ego[i];
        }
    }
    for (int i = tid; i < NB * NH; i += NTHREADS) lds_h[i] = 0.0f;
    asm volatile("s_wait_asynccnt 0" ::: "memory");  // async LDS writes visible
    __syncthreads();

    const int lane = tid & 31;
    const int wave = tid >> 5;   // 0..15
    const int lh   = lane >> 4;  // lane half: selects K sub-pair / D row offset
    const int l16  = lane & 15;  // N column within tile / M row for A

    for (int t = 0; t < NT; ++t) {
        // ---- gh = h @ Whh^T : 48 tiles of 16x16, 3 tiles per wave ----
        for (int tile = 0; tile < 3; ++tile) {
            int tIdx = wave * 3 + tile;          // 0..47
            int mt = tIdx / 24, nt = tIdx % 24;
            // A fragment base: row m = mt*16 + l16, K pair offset 2*lh
            const float* hrow = &lds_h[(mt * 16 + l16) * NH + 2 * lh];
            // B fragment base: pair p0 = lh, col n = nt*16 + l16
            const float* bcol = &lds_B[(lh * G3 + nt * 16 + l16) * 2];
            v8f acc = {};
#pragma unroll 4
            for (int kp = 0; kp < 32; ++kp) {    // K = 128 in steps of 4
                v2f a = *(const v2f*)(hrow + kp * 4);          // ds_load_b64
                v2f b = *(const v2f*)(bcol + kp * 4 * G3);     // ds_load_b64
                acc = __builtin_amdgcn_wmma_f32_16x16x4_f32(
                    false, a, false, b, (short)0, acc, false, false);
            }
            // D layout: VGPR v -> M = mt*16 + v + 8*lh, N = nt*16 + l16
            float* ghp = &lds_gh[(mt * 16 + 8 * lh) * G3 + nt * 16 + l16];
#pragma unroll
            for (int v = 0; v < 8; ++v) ghp[v * G3] = acc[v];
        }
        __syncthreads();

        // ---- gates + state update (gi computed on the fly, in_dim <= 4) ----
        for (int i = tid; i < NB * NH; i += NTHREADS) {
            int b = i >> 7, j = i & 127;
            const float* xv = &lds_x[(b * NT + t) * 4];
            float gr = lds_bih[j], gz = lds_bih[NH + j], gn = lds_bih[2 * NH + j];
            for (int d = 0; d < in_dim; ++d) {
                float xd = xv[d];
                gr += xd * lds_wih[j * 4 + d];
                gz += xd * lds_wih[(NH + j) * 4 + d];
                gn += xd * lds_wih[(2 * NH + j) * 4 + d];
            }
            float hr = lds_gh[b * G3 + j] + lds_bhh[j];
            float hz = lds_gh[b * G3 + NH + j] + lds_bhh[NH + j];
            float hn = lds_gh[b * G3 + 2 * NH + j] + lds_bhh[2 * NH + j];
            float r = 1.0f / (1.0f + __expf(-(gr + hr)));
            float z = 1.0f / (1.0f + __expf(-(gz + hz)));
            float n = tanhf(gn + r * hn);
            lds_h[i] = (1.0f - z) * n + z * lds_h[i];
        }
        __syncthreads();
    }

    float* wsg = ws + (size_t)g * (NB * NH);
    for (int i = tid; i < NB * NH; i += NTHREADS) wsg[i] = lds_h[i];
}

// Deterministic fixed-order combine (no float atomics): out = (b + f + e) / 3
__global__ void combine_kernel(const float* __restrict__ ws, float* __restrict__ out) {
    int i = blockIdx.x * blockDim.x + threadIdx.x;
    if (i < NB * NH) {
        out[i] = (ws[i] + ws[NB * NH + i] + ws[2 * NB * NH + i]) * (1.0f / 3.0f);
    }
}

extern "C" void kernel_launch(void* const* d_in, const int* in_sizes, int n_in,
                              void* d_out, int out_size, void* d_ws, size_t ws_size,
                              hipStream_t stream) {
    const float* bbox  = (const float*)d_in[0];
    const float* flow  = (const float*)d_in[1];
    const float* ego   = (const float*)d_in[2];
    const float* Wih_b = (const float*)d_in[3];
    const float* Whh_b = (const float*)d_in[4];
    const float* bih_b = (const float*)d_in[5];
    const float* bhh_b = (const float*)d_in[6];
    const float* Wih_f = (const float*)d_in[7];
    const float* Whh_f = (const float*)d_in[8];
    const float* bih_f = (const float*)d_in[9];
    const float* bhh_f = (const float*)d_in[10];
    const float* Wih_e = (const float*)d_in[11];
    const float* Whh_e = (const float*)d_in[12];
    const float* bih_e = (const float*)d_in[13];
    const float* bhh_e = (const float*)d_in[14];
    float* ws = (float*)d_ws;

    gru3_kernel<<<3, NTHREADS, 0, stream>>>(
        bbox, flow, ego,
        Wih_b, Whh_b, bih_b, bhh_b,
        Wih_f, Whh_f, bih_f, bhh_f,
        Wih_e, Whh_e, bih_e, bhh_e, ws);
    combine_kernel<<<(NB * NH + 255) / 256, 256, 0, stream>>>(ws, (float*)d_out);
}